// HybridQSTVAWithAnomaly_12558484373583
// MI455X (gfx1250) — compile-verified
//
#include <hip/hip_runtime.h>

#define T_ 10
#define D_ 8
#define L_ 2
#define B_ 32768
#define H_ 64
#define NREG 8   // 256 amplitudes / 32 lanes

typedef __attribute__((ext_vector_type(2))) float v2f;
typedef __attribute__((ext_vector_type(8))) float v8f;

__device__ __forceinline__ float shfx(float v, int m) {
    return __shfl_xor(v, m, 32);
}

// ---------------- quantum gates ----------------
// Amplitude index a in [0,256): lane = a & 31 (bits 0..4), reg = a >> 5 (bits 5..7).
// Qubit i acts on bit p = 7 - i.

__device__ __forceinline__ void gate_ry(float* re, float* im, int p, float c, float s, int lane) {
    if (p >= 5) {
        int m = 1 << (p - 5);
#pragma unroll
        for (int r = 0; r < NREG; ++r) {
            if (!(r & m)) {
                int r1 = r | m;
                float t0 = c * re[r] - s * re[r1];
                float t1 = s * re[r] + c * re[r1];
                re[r] = t0; re[r1] = t1;
                t0 = c * im[r] - s * im[r1];
                t1 = s * im[r] + c * im[r1];
                im[r] = t0; im[r1] = t1;
            }
        }
    } else {
        int mask = 1 << p;
        float sg = (lane & mask) ? s : -s;
#pragma unroll
        for (int r = 0; r < NREG; ++r) {
            float pr = shfx(re[r], mask);
            float pi = shfx(im[r], mask);
            re[r] = c * re[r] + sg * pr;
            im[r] = c * im[r] + sg * pi;
        }
    }
}

__device__ __forceinline__ void gate_rz(float* re, float* im, int p, float c, float s, int lane) {
    // diagonal: amp *= exp(-+ i*theta/2); bit0 -> (c,-s), bit1 -> (c,+s)
    if (p >= 5) {
        int m = 1 << (p - 5);
#pragma unroll
        for (int r = 0; r < NREG; ++r) {
            float sg = (r & m) ? -s : s;
            float nr = c * re[r] + sg * im[r];
            float ni = c * im[r] - sg * re[r];
            re[r] = nr; im[r] = ni;
        }
    } else {
        float sg = (lane & (1 << p)) ? -s : s;
#pragma unroll
        for (int r = 0; r < NREG; ++r) {
            float nr = c * re[r] + sg * im[r];
            float ni = c * im[r] - sg * re[r];
            re[r] = nr; im[r] = ni;
        }
    }
}

__device__ __forceinline__ void gate_rx(float* re, float* im, int p, float c, float s, int lane) {
    if (p >= 5) {
        int m = 1 << (p - 5);
#pragma unroll
        for (int r = 0; r < NREG; ++r) {
            if (!(r & m)) {
                int r1 = r | m;
                float n0r = c * re[r]  + s * im[r1];
                float n0i = c * im[r]  - s * re[r1];
                float n1r = c * re[r1] + s * im[r];
                float n1i = c * im[r1] - s * re[r];
                re[r] = n0r; im[r] = n0i; re[r1] = n1r; im[r1] = n1i;
            }
        }
    } else {
        int mask = 1 << p;
#pragma unroll
        for (int r = 0; r < NREG; ++r) {
            float pr = shfx(re[r], mask);
            float pi = shfx(im[r], mask);
            re[r] = c * re[r] + s * pi;   // symmetric for both halves
            im[r] = c * im[r] - s * pr;
        }
    }
}

__device__ __forceinline__ void gate_cnot(float* re, float* im, int pc, int pt, int lane) {
    if (pc >= 5 && pt >= 5) {                 // both register bits
        int mc = 1 << (pc - 5), mt = 1 << (pt - 5);
#pragma unroll
        for (int r = 0; r < NREG; ++r) {
            if ((r & mc) && !(r & mt)) {
                int r1 = r | mt;
                float t = re[r]; re[r] = re[r1]; re[r1] = t;
                t = im[r]; im[r] = im[r1]; im[r1] = t;
            }
        }
    } else if (pc >= 5) {                     // ctrl in regs, target across lanes
        int mc = 1 << (pc - 5), mask = 1 << pt;
#pragma unroll
        for (int r = 0; r < NREG; ++r) {
            if (r & mc) {
                re[r] = shfx(re[r], mask);
                im[r] = shfx(im[r], mask);
            }
        }
    } else if (pt >= 5) {                     // ctrl lane bit, target reg bit
        int mt = 1 << (pt - 5);
        bool hi = (lane >> pc) & 1;
#pragma unroll
        for (int r = 0; r < NREG; ++r) {
            if (!(r & mt)) {
                int r1 = r | mt;
                float a = re[r], b = re[r1];
                re[r]  = hi ? b : a;
                re[r1] = hi ? a : b;
                a = im[r]; b = im[r1];
                im[r]  = hi ? b : a;
                im[r1] = hi ? a : b;
            }
        }
    } else {                                  // both lane bits
        int mask = 1 << pt;
        bool hi = (lane >> pc) & 1;
#pragma unroll
        for (int r = 0; r < NREG; ++r) {
            float pr = shfx(re[r], mask);
            float pi = shfx(im[r], mask);
            re[r] = hi ? pr : re[r];
            im[r] = hi ? pi : im[r];
        }
    }
}

// ---------------- kernel 1: statevector sim, one sample per wave ----------------
__global__ __launch_bounds__(256) void qsim_kernel(const float* __restrict__ X,
                                                   const float* __restrict__ params,
                                                   float* __restrict__ qf) { // (B, T*D)
    int wave = (blockIdx.x * blockDim.x + threadIdx.x) >> 5;
    int lane = threadIdx.x & 31;
    const int N = T_ * B_;
    if (wave >= N) return;
    int t = wave / B_;
    int b = wave - t * B_;

    float re[NREG], im[NREG];
#pragma unroll
    for (int r = 0; r < NREG; ++r) { re[r] = 0.f; im[r] = 0.f; }
    if (lane == 0) re[0] = 1.f;   // |0...0>

    // data encoding: RY(x_i) then RZ(x_i) on qubit i
#pragma unroll
    for (int i = 0; i < D_; ++i) {
        float x = X[(size_t)wave * D_ + i];
        float s, c;
        __sincosf(0.5f * x, &s, &c);
        int p = 7 - i;
        gate_ry(re, im, p, c, s, lane);
        gate_rz(re, im, p, c, s, lane);
    }
    // variational layers
#pragma unroll
    for (int l = 0; l < L_; ++l) {
#pragma unroll
        for (int i = 0; i < D_; ++i) {
            int p = 7 - i;
            float s, c;
            __sincosf(0.5f * params[(l * D_ + i) * 3 + 0], &s, &c);
            gate_rx(re, im, p, c, s, lane);
            __sincosf(0.5f * params[(l * D_ + i) * 3 + 1], &s, &c);
            gate_ry(re, im, p, c, s, lane);
            __sincosf(0.5f * params[(l * D_ + i) * 3 + 2], &s, &c);
            gate_rz(re, im, p, c, s, lane);
        }
#pragma unroll
        for (int i = 0; i < D_; ++i) {
            int pc = 7 - i;
            int pt = 7 - ((i + 1) & 7);
            gate_cnot(re, im, pc, pt, lane);
        }
    }

    // <Z_i> expectations
    float prob[NREG], totp = 0.f;
#pragma unroll
    for (int r = 0; r < NREG; ++r) {
        prob[r] = re[r] * re[r] + im[r] * im[r];
        totp += prob[r];
    }
    float z[D_];
    z[0] = z[1] = z[2] = 0.f;
#pragma unroll
    for (int r = 0; r < NREG; ++r) {         // qubits 0,1,2 -> reg bits 2,1,0
        z[0] += (r & 4) ? -prob[r] : prob[r];
        z[1] += (r & 2) ? -prob[r] : prob[r];
        z[2] += (r & 1) ? -prob[r] : prob[r];
    }
#pragma unroll
    for (int i = 3; i < D_; ++i) {           // qubits 3..7 -> lane bits 4..0
        int p = 7 - i;
        z[i] = (lane & (1 << p)) ? -totp : totp;
    }
#pragma unroll
    for (int m = 1; m < 32; m <<= 1) {
#pragma unroll
        for (int i = 0; i < D_; ++i) z[i] += shfx(z[i], m);
    }
    if (lane == 0) {
        float* o = qf + (size_t)b * (T_ * D_) + t * D_;
#pragma unroll
        for (int i = 0; i < D_; ++i) o[i] = z[i];
    }
}

// ---------------- kernel 2: WMMA GEMM + LayerNorm + ReLU + head ----------------
// One wave computes h[16 rows][64 cols] via V_WMMA_F32_16X16X4_F32.
// C-layout: acc[c][v] at lane L = h[v + 8*(L>=16)][c*16 + (L&15)].
__global__ __launch_bounds__(256) void head_kernel(const float* __restrict__ qf,   // (B, 80)
                                                   const float* __restrict__ W1,   // (80, 64)
                                                   const float* __restrict__ b1,
                                                   const float* __restrict__ gamma,
                                                   const float* __restrict__ beta,
                                                   const float* __restrict__ W2,   // (64, 1)
                                                   const float* __restrict__ b2,
                                                   float* __restrict__ score) {    // (B)
    const int K = T_ * D_;  // 80
    int wave = blockIdx.x * (blockDim.x >> 5) + (threadIdx.x >> 5);
    int lane = threadIdx.x & 31;
    int half = lane >> 4;
    int lm   = lane & 15;
    int row0 = wave * 16;
    if (row0 >= B_) return;

    v8f acc[4];
#pragma unroll
    for (int c = 0; c < 4; ++c) {
#pragma unroll
        for (int v = 0; v < 8; ++v) acc[c][v] = 0.f;
    }

    const float* arow = qf + (size_t)(row0 + lm) * K;
#pragma unroll
    for (int k0 = 0; k0 < K; k0 += 4) {
        // A 16x4 f32 layout: lanes 0-15 -> K = k0+0/k0+1, lanes 16-31 -> K = k0+2/k0+3
        v2f a;
        a.x = arow[k0 + 2 * half];
        a.y = arow[k0 + 2 * half + 1];
#pragma unroll
        for (int c = 0; c < 4; ++c) {
            // B 4x16 f32 layout: VGPR0 = rows K0 (lanes 0-15) / K2 (lanes 16-31), VGPR1 = K1/K3
            v2f bm;
            bm.x = W1[(size_t)(k0 + 2 * half) * H_ + c * 16 + lm];
            bm.y = W1[(size_t)(k0 + 2 * half + 1) * H_ + c * 16 + lm];
            acc[c] = __builtin_amdgcn_wmma_f32_16x16x4_f32(
                false, a, false, bm, (short)0, acc[c], false, false);
        }
    }

    // + b1
    float h[4][8];
#pragma unroll
    for (int c = 0; c < 4; ++c) {
        float bb = b1[c * 16 + lm];
#pragma unroll
        for (int v = 0; v < 8; ++v) h[c][v] = acc[c][v] + bb;
    }

    // LayerNorm over 64 cols of each row: reduce across the 16-lane half-wave
    float mu[8];
#pragma unroll
    for (int v = 0; v < 8; ++v) mu[v] = h[0][v] + h[1][v] + h[2][v] + h[3][v];
#pragma unroll
    for (int m = 1; m < 16; m <<= 1) {
#pragma unroll
        for (int v = 0; v < 8; ++v) mu[v] += shfx(mu[v], m);
    }
#pragma unroll
    for (int v = 0; v < 8; ++v) mu[v] *= (1.f / 64.f);

    float var[8];
#pragma unroll
    for (int v = 0; v < 8; ++v) {
        float sq = 0.f;
#pragma unroll
        for (int c = 0; c < 4; ++c) { float d = h[c][v] - mu[v]; sq += d * d; }
        var[v] = sq;
    }
#pragma unroll
    for (int m = 1; m < 16; m <<= 1) {
#pragma unroll
        for (int v = 0; v < 8; ++v) var[v] += shfx(var[v], m);
    }

    float g[4], be[4], w2[4];
#pragma unroll
    for (int c = 0; c < 4; ++c) {
        g[c]  = gamma[c * 16 + lm];
        be[c] = beta [c * 16 + lm];
        w2[c] = W2   [c * 16 + lm];
    }

    float out[8];
#pragma unroll
    for (int v = 0; v < 8; ++v) {
        float rstd = rsqrtf(var[v] * (1.f / 64.f) + 1e-5f);
        float a2 = 0.f;
#pragma unroll
        for (int c = 0; c < 4; ++c) {
            float hn = (h[c][v] - mu[v]) * rstd * g[c] + be[c];
            hn = fmaxf(hn, 0.f);
            a2 += hn * w2[c];
        }
        out[v] = a2;
    }
#pragma unroll
    for (int m = 1; m < 16; m <<= 1) {
#pragma unroll
        for (int v = 0; v < 8; ++v) out[v] += shfx(out[v], m);
    }
    if (lm == 0) {
        float bias2 = b2[0];
#pragma unroll
        for (int v = 0; v < 8; ++v) {
            float sc = out[v] + bias2;
            score[row0 + half * 8 + v] = 1.f / (1.f + __expf(-sc));
        }
    }
}

extern "C" void kernel_launch(void* const* d_in, const int* in_sizes, int n_in,
                              void* d_out, int out_size, void* d_ws, size_t ws_size,
                              hipStream_t stream) {
    const float* X      = (const float*)d_in[0];
    const float* params = (const float*)d_in[1];
    const float* W1     = (const float*)d_in[2];
    const float* b1     = (const float*)d_in[3];
    const float* gamma  = (const float*)d_in[4];
    const float* beta   = (const float*)d_in[5];
    const float* W2     = (const float*)d_in[6];
    const float* b2     = (const float*)d_in[7];

    float* score = (float*)d_out;          // (B)
    float* qf    = score + B_;             // (B, T*D)

    const int N = T_ * B_;                 // 327680 samples, one wave each
    int blocks1 = (N + 7) / 8;             // 8 waves / 256-thread block
    qsim_kernel<<<blocks1, 256, 0, stream>>>(X, params, qf);

    int waves2  = B_ / 16;                 // 2048 row-blocks
    int blocks2 = (waves2 + 7) / 8;        // 8 waves / block
    head_kernel<<<blocks2, 256, 0, stream>>>(qf, W1, b1, gamma, beta, W2, b2, score);
}